// ScaffoldGNN_7825430413582
// MI455X (gfx1250) — compile-verified
//
#include <hip/hip_runtime.h>
#include <hip/hip_bf16.h>

typedef float v2f __attribute__((ext_vector_type(2)));
typedef float v8f __attribute__((ext_vector_type(8)));

#define NN 100000
#define NE 1600000
#define NG 2048
#define EPS_BN 1e-5f
#define K1P 96            // layer-1 K (74) padded to a multiple of 32

// ---------------- fills ----------------
__global__ void fill_f32(float* __restrict__ p, long long n, float v) {
    long long i = (long long)blockIdx.x * blockDim.x + threadIdx.x;
    if (i < n) p[i] = v;
}
__global__ void fill_u32(unsigned* __restrict__ p, long long n, unsigned v) {
    long long i = (long long)blockIdx.x * blockDim.x + threadIdx.x;
    if (i < n) p[i] = v;
}

// ---------------- padding for layer 1 ----------------
__global__ void pad_x96(const float* __restrict__ x, float* __restrict__ xp) {
    int idx = blockIdx.x * blockDim.x + threadIdx.x;          // NN*96
    if (idx >= NN * K1P) return;
    int i = idx / K1P, k = idx - i * K1P;
    xp[idx] = (k < 74) ? x[i * 74 + k] : 0.0f;
}
__global__ void pad_w96(const float* __restrict__ w, float* __restrict__ wp) {
    int idx = blockIdx.x * blockDim.x + threadIdx.x;          // 96*256
    if (idx >= K1P * 256) return;
    wp[idx] = (idx < 74 * 256) ? w[idx] : 0.0f;               // row-major: rows 74..95 zero
}

// ---------------- degree / dinv ----------------
__global__ void deg_kernel(const int* __restrict__ dst, float* __restrict__ deg) {
    int e = blockIdx.x * blockDim.x + threadIdx.x;
    if (e < NE) atomicAdd(&deg[dst[e]], 1.0f);
}
__global__ void dinv_kernel(float* __restrict__ deg) {
    int i = blockIdx.x * blockDim.x + threadIdx.x;
    if (i < NN) deg[i] = rsqrtf(deg[i] + 1.0f);               // self-loop: deg >= 1
}

// ---------------- f32 WMMA GEMM: C[M,N] = A[M,K] @ B[K,N] ----------------
// Requires M%32==0, K%32==0, N%64==0 (guaranteed by padding). 256 thr = 8 waves;
// wave tile 16x16; block tile 32(M) x 64(N); K staged 32-wide in LDS, all b128 staging.
#define AST 36   // mult of 4 (aligned b128 stores); 36*l mod 64 distinct for l=0..15
#define BST 72   // 2*72 mod 64 == 16 -> lane halves hit disjoint banks
__global__ __launch_bounds__(256) void gemm_wmma_f32(
    const float* __restrict__ A, const float* __restrict__ B, float* __restrict__ C,
    int M, int K, int N)
{
    __shared__ float As[32 * AST];
    __shared__ float Bs[32 * BST];
    const int tid   = threadIdx.x;
    const int wave  = tid >> 5;
    const int lane  = tid & 31;
    const int half  = lane >> 4;     // lane half selects K pair
    const int lidx  = lane & 15;     // M (for A) / N (for B,C) index within tile
    const int warpM = wave >> 2;     // 0..1
    const int warpN = wave & 3;      // 0..3
    const int blockM = blockIdx.x * 32;
    const int blockN = blockIdx.y * 64;

    // staging coordinates (guard-free, 16B aligned)
    const int arow = tid >> 3;            // 0..31
    const int ac0  = (tid & 7) * 4;       // 0..28
    const int bkk  = tid >> 4;            // 0..15
    const int bnn0 = (tid & 15) * 4;      // 0..60

    v8f acc = {};

    for (int k0 = 0; k0 < K; k0 += 32) {
        // stage A tile 32x32: one b128 per thread
        float4 av = *(const float4*)&A[(long long)(blockM + arow) * K + k0 + ac0];
        *(float4*)&As[arow * AST + ac0] = av;
        // stage B tile 32x64: two b128 per thread
        float4 bv0 = *(const float4*)&B[(long long)(k0 + bkk) * N + blockN + bnn0];
        float4 bv1 = *(const float4*)&B[(long long)(k0 + bkk + 16) * N + blockN + bnn0];
        *(float4*)&Bs[bkk * BST + bnn0] = bv0;
        *(float4*)&Bs[(bkk + 16) * BST + bnn0] = bv1;
        __syncthreads();

#pragma unroll
        for (int kk = 0; kk < 32; kk += 4) {
            // A frag (16x4 f32): lane(l<16): M=l, K=kk+{0,1}; lane(l>=16): M=l-16, K=kk+{2,3}
            const float* ap = &As[(warpM * 16 + lidx) * AST + kk + 2 * half];
            v2f a; a.x = ap[0]; a.y = ap[1];
            // B frag (4x16 f32): vgpr v holds K=kk+2*half+v, N=lidx
            v2f b;
            b.x = Bs[(kk + 2 * half + 0) * BST + warpN * 16 + lidx];
            b.y = Bs[(kk + 2 * half + 1) * BST + warpN * 16 + lidx];
            acc = __builtin_amdgcn_wmma_f32_16x16x4_f32(
                false, a, false, b, (short)0, acc, false, false);
        }
        __syncthreads();
    }

    // C layout: vgpr j -> (m = j + 8*half, n = lidx); no guards needed
    const int gn = blockN + warpN * 16 + lidx;
    const long long rowbase = (long long)(blockM + warpM * 16 + 8 * half) * N + gn;
#pragma unroll
    for (int j = 0; j < 8; ++j) {
        C[rowbase + (long long)j * N] = acc[j];
    }
}

// ---------------- aggregation ----------------
// agg[i,f] = xw[i,f] * dinv[i]^2   (self-loop term; full overwrite, no pre-zero needed)
__global__ void self_init4(const float4* __restrict__ xw, const float* __restrict__ dinv,
                           float4* __restrict__ agg, int F4) {
    int idx = blockIdx.x * blockDim.x + threadIdx.x;          // NN*F/4
    if (idx >= NN * F4) return;
    int i = idx / F4;
    float di = dinv[i];
    float s = di * di;
    float4 v = xw[idx];
    v.x *= s; v.y *= s; v.z *= s; v.w *= s;
    agg[idx] = v;
}

// edge-parallel scatter: thread = (edge, 4-float feature chunk); L2-resident atomics
__global__ void propagate(const float4* __restrict__ xw4, const int* __restrict__ src,
                          const int* __restrict__ dst, const float* __restrict__ dinv,
                          float* __restrict__ agg, int F4) {
    unsigned idx = blockIdx.x * blockDim.x + threadIdx.x;
    unsigned total = (unsigned)NE * (unsigned)F4;
    if (idx >= total) return;
    int e = idx / F4;
    int c = idx % F4;
    int s = src[e], d = dst[e];
    float nrm = dinv[s] * dinv[d];
    float4 v = xw4[(long long)s * F4 + c];
    float* o = agg + ((long long)d * F4 + c) * 4;
    atomicAdd(o + 0, v.x * nrm);
    atomicAdd(o + 1, v.y * nrm);
    atomicAdd(o + 2, v.z * nrm);
    atomicAdd(o + 3, v.w * nrm);
}

// h = relu( ((h + b) - rm) * rsqrt(rv+eps) * g + beta )   in place, float4
__global__ void bias_bn_relu4(float4* __restrict__ h, const float4* __restrict__ b,
                              const float4* __restrict__ g, const float4* __restrict__ beta,
                              const float4* __restrict__ rm, const float4* __restrict__ rv,
                              int F4) {
    int idx = blockIdx.x * blockDim.x + threadIdx.x;          // NN*F/4
    if (idx >= NN * F4) return;
    int f = idx % F4;
    float4 v = h[idx], vb = b[f], vg = g[f], vbe = beta[f], vm = rm[f], vv = rv[f];
    float4 r;
    r.x = fmaxf((v.x + vb.x - vm.x) * rsqrtf(vv.x + EPS_BN) * vg.x + vbe.x, 0.0f);
    r.y = fmaxf((v.y + vb.y - vm.y) * rsqrtf(vv.y + EPS_BN) * vg.y + vbe.y, 0.0f);
    r.z = fmaxf((v.z + vb.z - vm.z) * rsqrtf(vv.z + EPS_BN) * vg.z + vbe.z, 0.0f);
    r.w = fmaxf((v.w + vb.w - vm.w) * rsqrtf(vv.w + EPS_BN) * vg.w + vbe.w, 0.0f);
    h[idx] = r;
}

// ---------------- pooling ----------------
__device__ __forceinline__ unsigned fmap(float f) {
    unsigned u = __float_as_uint(f);
    return (u & 0x80000000u) ? ~u : (u | 0x80000000u);        // monotonic float->uint
}
__device__ __forceinline__ float funmap(unsigned u) {
    return __uint_as_float((u & 0x80000000u) ? (u & 0x7fffffffu) : ~u);
}

__global__ void cnt_kernel(const int* __restrict__ batch, float* __restrict__ cnt) {
    int i = blockIdx.x * blockDim.x + threadIdx.x;
    if (i < NN) atomicAdd(&cnt[batch[i]], 1.0f);
}
__global__ void pool_acc4(const float4* __restrict__ h3, const float4* __restrict__ b3,
                          const int* __restrict__ batch, float* __restrict__ sums,
                          unsigned* __restrict__ maxb, int F4) {
    int idx = blockIdx.x * blockDim.x + threadIdx.x;          // NN*F/4
    if (idx >= NN * F4) return;
    int i = idx / F4;
    int f = idx % F4;
    int gph = batch[i];
    float4 v = h3[idx], vb = b3[f];
    v.x += vb.x; v.y += vb.y; v.z += vb.z; v.w += vb.w;
    long long base = ((long long)gph * F4 + f) * 4;
    atomicAdd(&sums[base + 0], v.x);  atomicMax(&maxb[base + 0], fmap(v.x));
    atomicAdd(&sums[base + 1], v.y);  atomicMax(&maxb[base + 1], fmap(v.y));
    atomicAdd(&sums[base + 2], v.z);  atomicMax(&maxb[base + 2], fmap(v.z));
    atomicAdd(&sums[base + 3], v.w);  atomicMax(&maxb[base + 3], fmap(v.w));
}
__global__ void pool_final4(const float4* __restrict__ sums, const uint4* __restrict__ maxb,
                            const float* __restrict__ cnt, float4* __restrict__ out, int F4) {
    int idx = blockIdx.x * blockDim.x + threadIdx.x;          // NG*F/4
    if (idx >= NG * F4) return;
    int gph = idx / F4;
    float inv = 1.0f / fmaxf(cnt[gph], 1.0f);
    float4 s = sums[idx];
    uint4 m = maxb[idx];
    float4 r;
    r.x = s.x * inv + funmap(m.x);
    r.y = s.y * inv + funmap(m.y);
    r.z = s.z * inv + funmap(m.z);
    r.w = s.w * inv + funmap(m.w);
    out[idx] = r;
}

// ---------------- orchestration ----------------
extern "C" void kernel_launch(void* const* d_in, const int* in_sizes, int n_in,
                              void* d_out, int out_size, void* d_ws, size_t ws_size,
                              hipStream_t stream) {
    (void)in_sizes; (void)n_in; (void)out_size; (void)ws_size;
    const float* x    = (const float*)d_in[0];
    const int*   ei   = (const int*)d_in[1];
    const int*   src  = ei;
    const int*   dst  = ei + NE;
    const int*   batch= (const int*)d_in[2];
    const float* W1 = (const float*)d_in[3];  const float* b1  = (const float*)d_in[4];
    const float* g1 = (const float*)d_in[5];  const float* be1 = (const float*)d_in[6];
    const float* rm1= (const float*)d_in[7];  const float* rv1 = (const float*)d_in[8];
    const float* W2 = (const float*)d_in[9];  const float* b2  = (const float*)d_in[10];
    const float* g2 = (const float*)d_in[11]; const float* be2 = (const float*)d_in[12];
    const float* rm2= (const float*)d_in[13]; const float* rv2 = (const float*)d_in[14];
    const float* W3 = (const float*)d_in[15]; const float* b3  = (const float*)d_in[16];
    float* out = (float*)d_out;

    float* bufA = (float*)d_ws;                         // NN*384
    float* bufB = bufA + (long long)NN * 384;           // NN*384
    float* dinv = bufB + (long long)NN * 384;           // NN (deg -> dinv in place)
    float* sums = dinv + NN;                            // NG*384
    float* cnt  = sums + (long long)NG * 384;           // NG
    unsigned* maxb = (unsigned*)(cnt + NG);             // NG*384
    float* xpad = (float*)(maxb + (long long)NG * 384); // NN*96
    float* wpad = xpad + (long long)NN * K1P;           // 96*256

    auto gb = [](long long n) { return (unsigned)((n + 255) / 256); };

    // padded layer-1 operands (makes every GEMM guard-free)
    pad_x96<<<gb((long long)NN * K1P), 256, 0, stream>>>(x, xpad);
    pad_w96<<<gb(K1P * 256), 256, 0, stream>>>(W1, wpad);

    // degrees -> dinv
    fill_f32<<<gb(NN), 256, 0, stream>>>(dinv, NN, 0.0f);
    deg_kernel<<<gb(NE), 256, 0, stream>>>(dst, dinv);
    dinv_kernel<<<gb(NN), 256, 0, stream>>>(dinv);

    // ---- layer 1: 74 (padded 96) -> 256 ----
    gemm_wmma_f32<<<dim3(NN / 32, 256 / 64), 256, 0, stream>>>(xpad, wpad, bufB, NN, K1P, 256);
    self_init4<<<gb((long long)NN * 64), 256, 0, stream>>>((const float4*)bufB, dinv, (float4*)bufA, 64);
    propagate<<<gb((long long)NE * 64), 256, 0, stream>>>((const float4*)bufB, src, dst, dinv, bufA, 64);
    bias_bn_relu4<<<gb((long long)NN * 64), 256, 0, stream>>>((float4*)bufA, (const float4*)b1,
        (const float4*)g1, (const float4*)be1, (const float4*)rm1, (const float4*)rv1, 64);

    // ---- layer 2: 256 -> 256 ----
    gemm_wmma_f32<<<dim3(NN / 32, 256 / 64), 256, 0, stream>>>(bufA, W2, bufB, NN, 256, 256);
    self_init4<<<gb((long long)NN * 64), 256, 0, stream>>>((const float4*)bufB, dinv, (float4*)bufA, 64);
    propagate<<<gb((long long)NE * 64), 256, 0, stream>>>((const float4*)bufB, src, dst, dinv, bufA, 64);
    bias_bn_relu4<<<gb((long long)NN * 64), 256, 0, stream>>>((float4*)bufA, (const float4*)b2,
        (const float4*)g2, (const float4*)be2, (const float4*)rm2, (const float4*)rv2, 64);

    // ---- layer 3: 256 -> 384 (no BN/ReLU; bias folded into pooling) ----
    gemm_wmma_f32<<<dim3(NN / 32, 384 / 64), 256, 0, stream>>>(bufA, W3, bufB, NN, 256, 384);
    self_init4<<<gb((long long)NN * 96), 256, 0, stream>>>((const float4*)bufB, dinv, (float4*)bufA, 96);
    propagate<<<gb((long long)NE * 96), 256, 0, stream>>>((const float4*)bufB, src, dst, dinv, bufA, 96);

    // ---- pooling: mean + max over batch segments ----
    fill_f32<<<gb((long long)NG * 384), 256, 0, stream>>>(sums, (long long)NG * 384, 0.0f);
    fill_f32<<<gb(NG), 256, 0, stream>>>(cnt, NG, 0.0f);
    fill_u32<<<gb((long long)NG * 384), 256, 0, stream>>>(maxb, (long long)NG * 384, 0x007FFFFFu); // fmap(-inf)
    cnt_kernel<<<gb(NN), 256, 0, stream>>>(batch, cnt);
    pool_acc4<<<gb((long long)NN * 96), 256, 0, stream>>>((const float4*)bufA, (const float4*)b3,
        batch, sums, maxb, 96);
    pool_final4<<<gb((long long)NG * 96), 256, 0, stream>>>((const float4*)sums, (const uint4*)maxb,
        cnt, (float4*)out, 96);
}